// CustomCrossAttentionBaseline_38646115729786
// MI455X (gfx1250) — compile-verified
//
#include <hip/hip_runtime.h>
#include <hip/hip_bf16.h>
#include <stdint.h>

typedef __attribute__((ext_vector_type(2))) float v2f;
typedef __attribute__((ext_vector_type(8))) float v8f;

__device__ __forceinline__ v8f wmma4(v2f a, v2f b, v8f c) {
  // V_WMMA_F32_16X16X4_F32 : D(16x16 f32) = A(16x4 f32) * B(4x16 f32) + C
  return __builtin_amdgcn_wmma_f32_16x16x4_f32(false, a, false, b, (short)0, c, false, false);
}

#define NEGM (-3.4028234663852886e38f)
#define SCALE 0.15811388300841898f   // 40^-0.5

// ---------------- workspace layout (float offsets) ----------------
#define OFF_Q   ((size_t)0)                      // 64*4096*40 = 10,485,760  (reused as attn_out)
#define OFF_SIM ((size_t)10485760)               // 64*4096*80 = 20,971,520
#define OFF_K   ((size_t)31457280)               // 64*80*40   = 204,800 (j padded to 80, zeroed)
#define OFF_V   ((size_t)31662080)               // 204,800
#define OFF_RED ((size_t)31866880)               // 2 doubles (8B aligned)
#define OFF_WFS ((size_t)31866884)               // 1 float

// ---------------- CDNA5 async global->LDS helpers -----------------
__device__ __forceinline__ unsigned lds_off(const void* p) {
  // generic LDS pointer: addr[31:0] is the LDS byte offset (ISA flat aperture rule)
  return (unsigned)(unsigned long long)(uintptr_t)p;
}
__device__ __forceinline__ void gl_async_b128(unsigned lds_byte, const float* g) {
  asm volatile("global_load_async_to_lds_b128 %0, %1, off"
               :: "v"(lds_byte), "v"(g) : "memory");
}
template <int N>
__device__ __forceinline__ void wait_async() {
#if __has_builtin(__builtin_amdgcn_s_wait_asynccnt)
  __builtin_amdgcn_s_wait_asynccnt(N);
#else
  asm volatile("s_wait_asynccnt %0" :: "i"(N) : "memory");
#endif
}

// ------------------------------------------------------------------
__global__ void zero_kernel(float* __restrict__ p, long long n) {
  long long i = (long long)blockIdx.x * blockDim.x + threadIdx.x;
  long long stride = (long long)gridDim.x * blockDim.x;
  for (; i < n; i += stride) p[i] = 0.0f;
}

// ------------------------------------------------------------------
// Stage one 32-K chunk (A: 128x32, B: 32x64) into LDS via async copies.
// Exactly 6 async b128 per thread (uniform -> s_wait_asynccnt is exact).
__device__ __forceinline__ void stage_chunk(const float* __restrict__ A,
                                            const float* __restrict__ B,
                                            int M, int K, int row0, int col0,
                                            int kc0, float* asbuf, float* bsbuf,
                                            int t) {
#pragma unroll
  for (int it = 0; it < 4; ++it) {            // A: 1024 float4
    int id = t + it * 256;
    int r = id >> 3, c4 = (id & 7) * 4;
    int gr = row0 + r;
    if (gr >= M) gr = M - 1;                  // clamp: rows >= M are dead in epilogue
    gl_async_b128(lds_off(asbuf + r * 36 + c4), A + (size_t)gr * K + kc0 + c4);
  }
#pragma unroll
  for (int it = 0; it < 2; ++it) {            // B: 512 float4
    int id = t + it * 256;
    int r = id >> 4, c4 = (id & 15) * 4;
    gl_async_b128(lds_off(bsbuf + r * 72 + c4), B + (size_t)(kc0 + r) * 320 + col0 + c4);
  }
}

// C[M,320] = A[M,K] * B[K,320] with WMMA f32 16x16x4.
// 128x64 block tile, 8 waves x (2M x 2N) 16x16 subtiles, async double-buffered LDS.
// mode 0: split-head Q epilogue   (rows: b*4096+n  -> q[(b*8+h)*4096+n][d])
// mode 1: O-proj epilogue         (C[R*320+c] = val + bias[c])
// mode 2: split-head K/V epilogue (rows: b*77+j    -> kv[(b*8+h)*80+j][d])
__global__ __launch_bounds__(256) void gemm_wmma(
    const float* __restrict__ A, const float* __restrict__ B,
    float* __restrict__ C, const float* __restrict__ bias,
    int M, int K, int mode) {
  __shared__ __align__(16) float As[2][128 * 36];  // stride 36: conflict-free A-fragment reads
  __shared__ __align__(16) float Bs[2][32 * 72];   // stride 72: conflict-free B-fragment reads
  const int row0 = blockIdx.x * 128;
  const int col0 = blockIdx.y * 64;
  const int t = threadIdx.x;
  const int lane = t & 31, wave = t >> 5;
  const int wm = wave >> 1, wn = wave & 1;          // 4 (M) x 2 (N) waves
  const int lrow = lane & 15, lkg = (lane >> 4) * 2, lhi = (lane >> 4) * 8;

  v8f acc[2][2] = {{{}, {}}, {{}, {}}};

  const int nch = K >> 5;
  stage_chunk(A, B, M, K, row0, col0, 0, As[0], Bs[0], t);   // prologue

  for (int c = 0; c < nch; ++c) {
    const int cur = c & 1;
    if (c + 1 < nch) {
      stage_chunk(A, B, M, K, row0, col0, (c + 1) * 32, As[cur ^ 1], Bs[cur ^ 1], t);
      wait_async<6>();                      // prev chunk's 6 done (in-order completion)
    } else {
      wait_async<0>();
    }
    __syncthreads();                        // all waves' async data landed

    const float* as = As[cur];
    const float* bs = Bs[cur];
#pragma unroll
    for (int kk = 0; kk < 8; ++kk) {
      int kb = kk * 4 + lkg;
      v2f a0, a1, b0, b1;
      a0.x = as[(wm * 32 + lrow) * 36 + kb];      a0.y = as[(wm * 32 + lrow) * 36 + kb + 1];
      a1.x = as[(wm * 32 + 16 + lrow) * 36 + kb]; a1.y = as[(wm * 32 + 16 + lrow) * 36 + kb + 1];
      b0.x = bs[kb * 72 + wn * 32 + lrow];        b0.y = bs[(kb + 1) * 72 + wn * 32 + lrow];
      b1.x = bs[kb * 72 + wn * 32 + 16 + lrow];   b1.y = bs[(kb + 1) * 72 + wn * 32 + 16 + lrow];
      acc[0][0] = wmma4(a0, b0, acc[0][0]);
      acc[0][1] = wmma4(a0, b1, acc[0][1]);
      acc[1][0] = wmma4(a1, b0, acc[1][0]);
      acc[1][1] = wmma4(a1, b1, acc[1][1]);
    }
    __syncthreads();                        // safe to overwrite buf `cur` next iter
  }

  // epilogue: vgpr vr holds rows vr (lanes 0-15) / vr+8 (lanes 16-31)
#pragma unroll
  for (int ti = 0; ti < 2; ++ti) {
#pragma unroll
    for (int tj = 0; tj < 2; ++tj) {
      int cc = col0 + wn * 32 + tj * 16 + lrow;
#pragma unroll
      for (int vr = 0; vr < 8; ++vr) {
        int R = row0 + wm * 32 + ti * 16 + lhi + vr;
        if (R >= M) continue;
        float val = acc[ti][tj][vr];
        if (mode == 0) {                      // Q: R = b*4096 + n
          int b = R >> 12, nn = R & 4095;
          int h = cc / 40, d = cc - h * 40;
          C[(((size_t)(b * 8 + h)) * 4096 + nn) * 40 + d] = val;
        } else if (mode == 1) {               // O-proj + bias
          C[(size_t)R * 320 + cc] = val + bias[cc];
        } else {                              // K/V: R = b*77 + j, pad j->80
          int b = R / 77, j = R - b * 77;
          int h = cc / 40, d = cc - h * 40;
          C[(((size_t)(b * 8 + h)) * 80 + j) * 40 + d] = val;
        }
      }
    }
  }
}

// ------------------------------------------------------------------
// Pass 1: sim = q k^T per (bh, 64-row tile), store sim, accumulate masked
// fp64 sum/sumsq for the global std.
__global__ __launch_bounds__(320) void sim_kernel(
    const float* __restrict__ q, const float* __restrict__ kbuf,
    const int* __restrict__ capt, float* __restrict__ sim,
    double* __restrict__ red) {
  __shared__ float Qs[64 * 44];
  __shared__ float Ks[80 * 44];
  __shared__ int   km[80];
  __shared__ double sred[320], sred2[320];
  const int bh = blockIdx.y, b = bh >> 3;
  const int row0 = blockIdx.x * 64;
  const int t = threadIdx.x;

  for (int id = t; id < 64 * 40; id += 320) {
    int r = id / 40, c = id - r * 40;
    Qs[r * 44 + c] = q[(size_t)bh * 163840 + (size_t)(row0 + r) * 40 + c];
  }
  for (int id = t; id < 80 * 40; id += 320) {
    int j = id / 40, c = id - j * 40;
    Ks[j * 44 + c] = kbuf[(size_t)bh * 3200 + j * 40 + c];  // rows 77..79 pre-zeroed
  }
  if (t < 80) km[t] = (t < 77) && (capt[b * 77 + t] >= 0);
  __syncthreads();

  const int lane = t & 31, wave = t >> 5;
  const int lrow = lane & 15, lkg = (lane >> 4) * 2, lhi = (lane >> 4) * 8;
  double lsum = 0.0, lsq = 0.0;

  // 20 subtiles (4 M x 5 J) over 10 waves: 2 each
#pragma unroll
  for (int si = 0; si < 2; ++si) {
    int s = wave * 2 + si;
    int mi = s / 5, ni = s - mi * 5;
    v8f acc = {};
#pragma unroll
    for (int kk = 0; kk < 10; ++kk) {       // K = d_head = 40
      int kb = kk * 4 + lkg;
      v2f a;  a.x = Qs[(mi * 16 + lrow) * 44 + kb]; a.y = Qs[(mi * 16 + lrow) * 44 + kb + 1];
      v2f bb; bb.x = Ks[(ni * 16 + lrow) * 44 + kb]; bb.y = Ks[(ni * 16 + lrow) * 44 + kb + 1];
      acc = wmma4(a, bb, acc);
    }
    int jc = ni * 16 + lrow;
    int mv = km[jc];
#pragma unroll
    for (int vr = 0; vr < 8; ++vr) {
      float val = acc[vr];
      int r = mi * 16 + lhi + vr;
      sim[(size_t)bh * 327680 + (size_t)(row0 + r) * 80 + jc] = val;
      if (mv) { lsum += (double)val; lsq += (double)val * (double)val; }
    }
  }
  sred[t] = lsum; sred2[t] = lsq;
  __syncthreads();
  if (t == 0) {
    double s = 0.0, s2 = 0.0;
    for (int i = 0; i < 320; ++i) { s += sred[i]; s2 += sred2[i]; }
    atomicAdd(&red[0], s);                  // global_atomic_add_f64
    atomicAdd(&red[1], s2);
  }
}

// ------------------------------------------------------------------
__global__ void finalize_kernel(const int* __restrict__ capt,
                                const float* __restrict__ strength,
                                const double* __restrict__ red,
                                float* __restrict__ wfs) {
  if (threadIdx.x == 0 && blockIdx.x == 0) {
    long long c = 0;
    for (int i = 0; i < 8 * 77; ++i) c += (capt[i] >= 0) ? 1 : 0;
    double cnt = (double)c * 4096.0 * 8.0;  // broadcast over n and heads
    double mean = red[0] / cnt;
    double var = (red[1] - cnt * mean * mean) / (cnt - 1.0);
    if (var < 0.0) var = 0.0;
    *wfs = (float)(sqrt(var) * (double)strength[0]);
  }
}

// ------------------------------------------------------------------
// Pass 2: mask + wf + scale + softmax, then attn(64x80) @ V(80x40) via WMMA.
__global__ __launch_bounds__(384) void attn_kernel(
    const float* __restrict__ sim, const float* __restrict__ vbuf,
    const float* __restrict__ cam, const int* __restrict__ capt,
    const unsigned char* __restrict__ gpm, const float* __restrict__ wfs_p,
    float* __restrict__ attn_out) {
  __shared__ float Ps[64 * 84];
  __shared__ float Vs[80 * 52];
  __shared__ int   km[80], gm[80];
  const int bh = blockIdx.y, b = bh >> 3, h = bh & 7;
  const int row0 = blockIdx.x * 64;
  const int t = threadIdx.x;

  if (t < 80) {
    km[t] = (t < 77) && (capt[b * 77 + t] >= 0);
    gm[t] = (t < 77) && (gpm[b * 77 + t] != 0);
  }
  for (int id = t; id < 80 * 52; id += 384) {
    int j = id / 52, d = id - j * 52;
    Vs[id] = (d < 40) ? vbuf[(size_t)bh * 3200 + j * 40 + d] : 0.0f;
  }
  for (int id = t; id < 64 * 80; id += 384) {
    int r = id / 80, j = id - r * 80;
    Ps[r * 84 + j] = sim[(size_t)bh * 327680 + (size_t)(row0 + r) * 80 + j];
  }
  __syncthreads();

  const float wfs = *wfs_p;
  if (t < 64) {
    const int r = t;
    const float* camrow = cam + ((size_t)b * 4096 + row0 + r) * 77;
    float mx = -INFINITY;
    for (int j = 0; j < 80; ++j) {
      float wf = gm[j] ? camrow[j] * wfs : NEGM;
      float s = (km[j] ? Ps[r * 84 + j] : NEGM) + wf;   // pad cols -> -inf
      float z = s * SCALE;
      Ps[r * 84 + j] = z;
      mx = fmaxf(mx, z);
    }
    float sum = 0.0f;
    for (int j = 0; j < 80; ++j) {
      float e = __expf(Ps[r * 84 + j] - mx);
      Ps[r * 84 + j] = e;
      sum += e;
    }
    float inv = 1.0f / sum;
    for (int j = 0; j < 80; ++j) Ps[r * 84 + j] *= inv;
  }
  __syncthreads();

  const int lane = t & 31, wave = t >> 5;
  const int lrow = lane & 15, lkg = (lane >> 4) * 2, lhi = (lane >> 4) * 8;
  const int mi = wave / 3, ni = wave - mi * 3;          // 4 x 3 subtiles, 12 waves
  v8f acc = {};
#pragma unroll
  for (int kk = 0; kk < 20; ++kk) {                     // K = 80 (padded j)
    int kb = kk * 4 + lkg;
    v2f a;  a.x = Ps[(mi * 16 + lrow) * 84 + kb]; a.y = Ps[(mi * 16 + lrow) * 84 + kb + 1];
    v2f bb; bb.x = Vs[kb * 52 + ni * 16 + lrow];  bb.y = Vs[(kb + 1) * 52 + ni * 16 + lrow];
    acc = wmma4(a, bb, acc);
  }
  int d = ni * 16 + lrow;
  if (d < 40) {
#pragma unroll
    for (int vr = 0; vr < 8; ++vr) {
      int n = row0 + mi * 16 + lhi + vr;
      attn_out[((size_t)b * 4096 + n) * 320 + h * 40 + d] = acc[vr];  // head-merged
    }
  }
}

// ------------------------------------------------------------------
extern "C" void kernel_launch(void* const* d_in, const int* in_sizes, int n_in,
                              void* d_out, int out_size, void* d_ws, size_t ws_size,
                              hipStream_t stream) {
  const float* x    = (const float*)d_in[0];   // [8,4096,320]
  const float* embs = (const float*)d_in[1];   // [8,77,768]
  const float* Wq   = (const float*)d_in[2];   // [320,320]
  const float* Wk   = (const float*)d_in[3];   // [768,320]
  const float* Wv   = (const float*)d_in[4];   // [768,320]
  const float* Wo   = (const float*)d_in[5];   // [320,320]
  const float* bo   = (const float*)d_in[6];   // [320]
  const float* cam  = (const float*)d_in[7];   // [8,4096,77]
  const float* strength = (const float*)d_in[8];             // [1]
  const int*   capt = (const int*)d_in[9];                   // [8,77]
  const unsigned char* gpm = (const unsigned char*)d_in[10]; // [8,77] bool
  float* out = (float*)d_out;
  float* ws  = (float*)d_ws;

  float*  qbuf = ws + OFF_Q;      // q, later reused as attn output (fully overwritten)
  float*  simb = ws + OFF_SIM;
  float*  kbuf = ws + OFF_K;
  float*  vbuf = ws + OFF_V;
  double* red  = (double*)(ws + OFF_RED);
  float*  wfs  = ws + OFF_WFS;

  // zero padded K/V rows + fp64 accumulators (every call: atomics accumulate)
  zero_kernel<<<256, 256, 0, stream>>>(kbuf, (long long)(204800 * 2 + 8));

  // projections (async double-buffered WMMA GEMMs)
  gemm_wmma<<<dim3(256, 5), 256, 0, stream>>>(x,    Wq, qbuf, nullptr, 32768, 320, 0);
  gemm_wmma<<<dim3(5, 5),   256, 0, stream>>>(embs, Wk, kbuf, nullptr,   616, 768, 2);
  gemm_wmma<<<dim3(5, 5),   256, 0, stream>>>(embs, Wv, vbuf, nullptr,   616, 768, 2);

  // sim + masked global moments
  sim_kernel<<<dim3(64, 64), 320, 0, stream>>>(qbuf, kbuf, capt, simb, red);
  finalize_kernel<<<1, 32, 0, stream>>>(capt, strength, red, wfs);

  // softmax + AV (writes into qbuf as head-merged [32768,320])
  attn_kernel<<<dim3(64, 64), 384, 0, stream>>>(simb, vbuf, cam, capt, gpm, wfs, qbuf);

  // output projection + bias
  gemm_wmma<<<dim3(256, 5), 256, 0, stream>>>(qbuf, Wo, out, bo, 32768, 320, 1);
}